// StatisticExtractor_670014898670
// MI455X (gfx1250) — compile-verified
//
#include <hip/hip_runtime.h>
#include <float.h>

// Problem shape (from reference): data [B,C,L] f32, pad_mask [B,L] i32.
#define Bdim 128
#define Cdim 32
#define Ldim 8192
#define WORDS (Ldim / 32)   // 256 bitmask words per batch row
#define NT 256              // threads per block (8 waves of 32)
#define NW 8                // waves per block

typedef float v2f __attribute__((ext_vector_type(2)));
typedef float v8f __attribute__((ext_vector_type(8)));

// -------------------------------------------------------------------------
// Kernel 1: pack pad_mask[b, 0..L) (int32) into a bitmask (1 bit/elem).
// Cuts mask traffic from 4MB x 32 channel-reloads to 128KB total.
// -------------------------------------------------------------------------
__global__ void __launch_bounds__(NT) pack_mask_kernel(const int* __restrict__ pm,
                                                       unsigned* __restrict__ bmg) {
  __shared__ unsigned sbm[WORDS];
  const int b = blockIdx.x;
  const int tid = threadIdx.x;
  sbm[tid] = 0u;
  __syncthreads();
  const int* row = pm + (size_t)b * Ldim;
  #pragma unroll
  for (int j = 0; j < Ldim / NT; ++j) {
    int idx = j * NT + tid;                  // coalesced
    if (row[idx] != 0) atomicOr(&sbm[idx >> 5], 1u << (idx & 31));
  }
  __syncthreads();
  bmg[(size_t)b * WORDS + tid] = sbm[tid];
}

__device__ __forceinline__ float cleanf(float x) {
  // nan_to_num(nan=0, posinf=0, neginf=0)
  return (x == x && x <= FLT_MAX && x >= -FLT_MAX) ? x : 0.0f;
}

// Rank-locate over a shared histogram: find bin containing the k-th element
// (0-indexed) and residual rank. Wave 0 does a shuffle-based prefix scan of
// the 256 per-thread segment sums (5 shuffle steps instead of a 256-long serial scan).
__device__ __forceinline__ void locate_rank(unsigned* hist, int nbins, int k,
                                            unsigned* segsum, unsigned* segbase,
                                            int* selBin, int* selRem, int tid) {
  const int seg = nbins >> 8;               // bins per thread
  const int base0 = tid * seg;
  unsigned s = 0;
  for (int j = 0; j < seg; ++j) s += hist[base0 + j];
  segsum[tid] = s;
  __syncthreads();
  if (tid < 32) {                           // wave-0 parallel scan, 8 seg/lane
    unsigned loc[8];
    unsigned tot = 0;
    #pragma unroll
    for (int j = 0; j < 8; ++j) { loc[j] = segsum[tid * 8 + j]; tot += loc[j]; }
    unsigned inc = tot;
    #pragma unroll
    for (int o = 1; o < 32; o <<= 1) {
      unsigned v = __shfl_up(inc, o);
      if (tid >= o) inc += v;
    }
    unsigned base = inc - tot;              // exclusive prefix across lanes
    #pragma unroll
    for (int j = 0; j < 8; ++j) { segbase[tid * 8 + j] = base; base += loc[j]; }
  }
  __syncthreads();
  unsigned uk = (unsigned)k;
  if (uk >= segbase[tid] && uk < segbase[tid] + s) {
    unsigned rr = uk - segbase[tid];
    for (int j = 0; j < seg; ++j) {
      unsigned h = hist[base0 + j];
      if (rr < h) { *selBin = base0 + j; *selRem = (int)rr; break; }
      rr -= h;
    }
  }
  __syncthreads();
}

// -------------------------------------------------------------------------
// Kernel 2: one workgroup per (b,c) row. The 32KB row is pulled into LDS by
// GLOBAL_LOAD_ASYNC_TO_LDS_B128 (ASYNCcnt path, no VGPR round trip), read
// exactly once from HBM (roofline: 128MB @ 23.3TB/s ~ 5.7us). Then:
//  - masked sums Sx,Sxx,Sd,Sdd accumulate on the matrix pipe via
//    V_WMMA_F32_16X16X4_F32 with B = ones (reduction-as-GEMV); min/max on
//    VALU co-executes with the WMMA chain. Loop is branch-free, uniform
//    trip count -> EXEC stays all-ones as WMMA requires.
//  - xs is converted in place to monotone sort keys (invalid -> 0xFFFFFFFF,
//    the reference's +fmax sentinel), then exact q25/q75 order statistics
//    come from a 3-level radix select (11+11+10 bits), with histogram
//    builds grouped across the 4 ranks (ranks are sorted, equal bins are
//    adjacent).
// -------------------------------------------------------------------------
__global__ void __launch_bounds__(NT) stats_kernel(const float* __restrict__ data,
                                                   const unsigned* __restrict__ bmg,
                                                   float* __restrict__ out) {
  __shared__ float xs[Ldim];          // 32 KB staged row (later reused as keys)
  __shared__ unsigned bmsh[WORDS];    // 1 KB bitmask
  __shared__ unsigned hist[2048];     // 8 KB radix histogram
  __shared__ unsigned segsum[NT];
  __shared__ unsigned segbase[NT];
  __shared__ float rSx[NW], rSxx[NW], rSd[NW], rSdd[NW], rMn[NW], rMx[NW];
  __shared__ int rN[NW], rCD[NW];
  __shared__ int ranks[4];
  __shared__ float qv[4];
  __shared__ int sBinA[4], sRemA[4], sBinB[4], sRemB[4];
  __shared__ int sB2, sR2;
  __shared__ int sValid;
  __shared__ float sF25, sF75;

  const int tid = threadIdx.x;
  const int lane = tid & 31;
  const int wv = tid >> 5;
  const int b = blockIdx.x >> 5;      // 4096 blocks = 128 b * 32 c
  const int c = blockIdx.x & 31;

  // ---- async-stage row into LDS (32 KB, 16B per lane per op) ----
  {
    const char* rowbase = (const char*)(data + (size_t)(b * Cdim + c) * Ldim);
    // low 32 bits of a generic pointer to LDS == wave-relative LDS address
    unsigned ldsbase = (unsigned)(size_t)(void*)xs;
    #pragma unroll
    for (int j = 0; j < Ldim / (4 * NT); ++j) {   // 8 iterations
      unsigned voff = (unsigned)((j * NT + tid) * 16);
      asm volatile("global_load_async_to_lds_b128 %0, %1, %2"
                   :: "v"(ldsbase + voff), "v"(voff), "s"(rowbase)
                   : "memory");
    }
  }
  bmsh[tid] = bmg[(size_t)b * WORDS + tid];
  asm volatile("s_wait_asynccnt 0" ::: "memory");
  __syncthreads();

  // ---- n (valid count) and cd (valid adjacent-pair count) from bitmask ----
  {
    unsigned w = bmsh[tid];
    int nn = __popc(w);
    unsigned nxt = (tid < WORDS - 1) ? bmsh[tid + 1] : 0u;
    int cc = __popc(w & (w >> 1)) + (int)((w >> 31) & nxt & 1u);
    for (int off = 16; off > 0; off >>= 1) {
      nn += __shfl_xor(nn, off);
      cc += __shfl_xor(cc, off);
    }
    if (lane == 0) { rN[wv] = nn; rCD[wv] = cc; }
  }

  // ---- main pass: 4 masked reductions on the matrix pipe + min/max ----
  v2f bones = {1.0f, 1.0f};           // B = ones => column-replicated row sums
  v8f aX = {}, aXX = {}, aD = {}, aDD = {};
  float vmn = FLT_MAX, vmx = -FLT_MAX;
  const unsigned char* bmb = (const unsigned char*)bmsh;  // byte view for b2
  for (int it = 0; it < (Ldim / 64) / NW; ++it) {  // uniform trip count: 16
    int ck = it * NW + wv;
    int e0 = ck * 64 + lane * 2;
    int e1 = e0 + 1;
    unsigned inb = (unsigned)(e0 + 2 < Ldim);
    int e2c = inb ? (e0 + 2) : (Ldim - 1);         // clamped, branch-free
    float x0 = xs[e0], x1 = xs[e1], x2 = xs[e2c];
    unsigned w = bmsh[e0 >> 5];                    // e0 even -> e0,e1 same word
    int sh = e0 & 31;
    unsigned b0 = (w >> sh) & 1u;
    unsigned b1 = (w >> (sh + 1)) & 1u;
    unsigned b2 = ((unsigned)(bmb[e2c >> 3] >> (e2c & 7)) & 1u) & inb;
    float a0 = b0 ? x0 : 0.0f;
    float a1 = b1 ? x1 : 0.0f;
    float dd0 = (b0 & b1) ? (x1 - x0) : 0.0f;      // diff valid iff both ends
    float dd1 = (b1 & b2) ? (x2 - x1) : 0.0f;
    vmn = fminf(vmn, b0 ? x0 : FLT_MAX);
    vmn = fminf(vmn, b1 ? x1 : FLT_MAX);
    vmx = fmaxf(vmx, b0 ? x0 : -FLT_MAX);
    vmx = fmaxf(vmx, b1 ? x1 : -FLT_MAX);
    v2f Ax  = {a0, a1};
    v2f Axx = {a0 * a0, a1 * a1};
    v2f Ad  = {dd0, dd1};
    v2f Add = {dd0 * dd0, dd1 * dd1};
    aX  = __builtin_amdgcn_wmma_f32_16x16x4_f32(false, Ax,  false, bones, (short)0, aX,  false, false);
    aXX = __builtin_amdgcn_wmma_f32_16x16x4_f32(false, Axx, false, bones, (short)0, aXX, false, false);
    aD  = __builtin_amdgcn_wmma_f32_16x16x4_f32(false, Ad,  false, bones, (short)0, aD,  false, false);
    aDD = __builtin_amdgcn_wmma_f32_16x16x4_f32(false, Add, false, bones, (short)0, aDD, false, false);
  }
  // fold 8 acc VGPRs per lane, then butterfly-reduce the wave
  float pX = 0, pXX = 0, pD = 0, pDD = 0;
  #pragma unroll
  for (int i = 0; i < 8; ++i) { pX += aX[i]; pXX += aXX[i]; pD += aD[i]; pDD += aDD[i]; }
  for (int off = 16; off > 0; off >>= 1) {
    pX  += __shfl_xor(pX, off);
    pXX += __shfl_xor(pXX, off);
    pD  += __shfl_xor(pD, off);
    pDD += __shfl_xor(pDD, off);
    vmn = fminf(vmn, __shfl_xor(vmn, off));
    vmx = fmaxf(vmx, __shfl_xor(vmx, off));
  }
  if (lane == 0) {
    rSx[wv] = pX; rSxx[wv] = pXX; rSd[wv] = pD; rSdd[wv] = pDD;
    rMn[wv] = vmn; rMx[wv] = vmx;
  }
  __syncthreads();   // moments pass done; xs free to be overwritten

  // ---- convert xs in place to monotone sort keys (all threads) ----
  unsigned* ks = (unsigned*)xs;
  #pragma unroll
  for (int j = 0; j < Ldim / NT; ++j) {     // uniform trip count: 32
    int i = j * NT + tid;
    unsigned valid = (bmsh[i >> 5] >> (i & 31)) & 1u;
    unsigned u = __float_as_uint(xs[i]);
    unsigned k = (u & 0x80000000u) ? ~u : (u | 0x80000000u);
    ks[i] = valid ? k : 0xFFFFFFFFu;        // sentinel == reference's +fmax
  }

  // ---- finalize moments (thread 0); 16 identical WMMA columns => /16 ----
  if (tid == 0) {
    float Sx = 0, Sxx = 0, Sd = 0, Sdd = 0, mn = FLT_MAX, mx = -FLT_MAX;
    int n = 0, cd = 0;
    for (int w = 0; w < NW; ++w) {
      Sx += rSx[w]; Sxx += rSxx[w]; Sd += rSd[w]; Sdd += rSdd[w];
      mn = fminf(mn, rMn[w]); mx = fmaxf(mx, rMx[w]);
      n += rN[w]; cd += rCD[w];
    }
    const float inv16 = 1.0f / 16.0f;
    Sx *= inv16; Sxx *= inv16; Sd *= inv16; Sdd *= inv16;
    float fn = (float)n, fcd = (float)cd;
    float mean  = Sx / fn;
    float stdv  = sqrtf(fmaxf(Sxx / fn - mean * mean, 0.0f));
    float dmean = Sd / fcd;
    float dstd  = sqrtf(fmaxf(Sdd / fcd - dmean * dmean, 0.0f));
    float rms   = sqrtf(Sxx / fn);
    size_t ob = (size_t)b * (Cdim * 10) + c;
    out[ob + 0 * Cdim] = cleanf(mean);
    out[ob + 1 * Cdim] = cleanf(stdv);
    out[ob + 2 * Cdim] = cleanf(mx);     // sentinel -FLT_MAX == finfo.min
    out[ob + 3 * Cdim] = cleanf(mn);     // sentinel +FLT_MAX == finfo.max
    out[ob + 6 * Cdim] = cleanf(dmean);
    out[ob + 7 * Cdim] = cleanf(dstd);
    out[ob + 8 * Cdim] = cleanf(Sxx);    // energy
    out[ob + 9 * Cdim] = cleanf(rms);
    sValid = (n > 0);
    if (n > 0) {
      float nm1 = (float)(n - 1);
      float p25 = 0.25f * nm1, p75 = 0.75f * nm1;
      int lo25 = (int)floorf(p25), hi25 = (int)ceilf(p25);
      int lo75 = (int)floorf(p75), hi75 = (int)ceilf(p75);
      ranks[0] = lo25; ranks[1] = hi25; ranks[2] = lo75; ranks[3] = hi75;
      sF25 = p25 - (float)lo25; sF75 = p75 - (float)lo75;
    } else {
      out[ob + 4 * Cdim] = 0.0f;         // reference: empty row -> 0
      out[ob + 5 * Cdim] = 0.0f;
    }
  }
  __syncthreads();
  if (!sValid) return;

  // ---- Level A: key[31:21] (2048 bins), one histogram for all 4 ranks ----
  for (int j = 0; j < 2048 / NT; ++j) hist[j * NT + tid] = 0u;
  __syncthreads();
  #pragma unroll 4
  for (int j = 0; j < Ldim / NT; ++j) {
    unsigned k = ks[j * NT + tid];
    atomicAdd(&hist[k >> 21], 1u);
  }
  __syncthreads();
  for (int r = 0; r < 4; ++r)
    locate_rank(hist, 2048, ranks[r], segsum, segbase, &sBinA[r], &sRemA[r], tid);

  // ---- Level B: key[20:10], histogram builds grouped by distinct bin A ----
  {
    int lastA = -1;
    for (int r = 0; r < 4; ++r) {
      unsigned binA = (unsigned)sBinA[r];     // uniform: all threads read shared
      if ((int)binA != lastA) {
        for (int j = 0; j < 2048 / NT; ++j) hist[j * NT + tid] = 0u;
        __syncthreads();
        #pragma unroll 4
        for (int j = 0; j < Ldim / NT; ++j) {
          unsigned k = ks[j * NT + tid];
          if ((k >> 21) == binA) atomicAdd(&hist[(k >> 10) & 0x7FFu], 1u);
        }
        __syncthreads();
        lastA = (int)binA;
      }
      locate_rank(hist, 2048, sRemA[r], segsum, segbase, &sBinB[r], &sRemB[r], tid);
    }
  }

  // ---- Level C: key[9:0], grouped by distinct (A,B); reconstruct value ----
  {
    unsigned lastAB = 0xFFFFFFFFu;
    for (int r = 0; r < 4; ++r) {
      unsigned binA = (unsigned)sBinA[r];
      unsigned binB = (unsigned)sBinB[r];
      unsigned topAB = (binA << 11) | binB;
      if (topAB != lastAB) {
        for (int j = 0; j < 1024 / NT; ++j) hist[j * NT + tid] = 0u;
        __syncthreads();
        #pragma unroll 4
        for (int j = 0; j < Ldim / NT; ++j) {
          unsigned k = ks[j * NT + tid];
          if ((k >> 10) == topAB) atomicAdd(&hist[k & 0x3FFu], 1u);
        }
        __syncthreads();
        lastAB = topAB;
      }
      locate_rank(hist, 1024, sRemB[r], segsum, segbase, &sB2, &sR2, tid);
      if (tid == 0) {
        unsigned qk = (binA << 21) | (binB << 10) | (unsigned)sB2;
        qv[r] = (qk & 0x80000000u) ? __uint_as_float(qk & 0x7FFFFFFFu)
                                   : __uint_as_float(~qk);
      }
      __syncthreads();
    }
  }

  if (tid == 0) {
    float q25 = qv[0] + sF25 * (qv[1] - qv[0]);
    float q75 = qv[2] + sF75 * (qv[3] - qv[2]);
    size_t ob = (size_t)b * (Cdim * 10) + c;
    out[ob + 4 * Cdim] = cleanf(q25);
    out[ob + 5 * Cdim] = cleanf(q75);
  }
}

// -------------------------------------------------------------------------
extern "C" void kernel_launch(void* const* d_in, const int* in_sizes, int n_in,
                              void* d_out, int out_size, void* d_ws, size_t ws_size,
                              hipStream_t stream) {
  const float* data = (const float*)d_in[0];   // [B,C,L] f32
  const int* pm = (const int*)d_in[1];         // [B,L] i32
  float* out = (float*)d_out;                  // [B, C*10] f32
  unsigned* bmg = (unsigned*)d_ws;             // 128*256 u32 = 128 KB bitmask

  pack_mask_kernel<<<Bdim, NT, 0, stream>>>(pm, bmg);
  stats_kernel<<<Bdim * Cdim, NT, 0, stream>>>(data, bmg, out);
}